// ProgLoss_37031208026603
// MI455X (gfx1250) — compile-verified
//
#include <hip/hip_runtime.h>
#include <math.h>
#include <stdint.h>

// ---- problem constants ----
#define S_ROWS 524288
#define D_COLS 63
#define ROWS_PER_TILE 64                 // 64 rows * 63 f32 = 16128 B per array per tile
#define TILE_ELEMS (ROWS_PER_TILE * D_COLS)   // 4032
#define TILES_PER_BLOCK 4
#define THREADS 64                       // 2 waves (wave32)
#define NBLOCKS (S_ROWS / (ROWS_PER_TILE * TILES_PER_BLOCK))  // 2048
#define NACC 37                          // 36 sums + 1 min (first cub row)
#define WS_STRIDE 40
#define POS_W 0.2f
#define BIGF 3.0e38f

typedef unsigned int u32x4 __attribute__((ext_vector_type(4)));
typedef int i32x8 __attribute__((ext_vector_type(8)));
typedef int i32x4 __attribute__((ext_vector_type(4)));

#if __has_builtin(__builtin_amdgcn_tensor_load_to_lds) && __has_builtin(__builtin_amdgcn_s_wait_tensorcnt)
#define USE_TDM 1
#else
#define USE_TDM 0
#endif

// ---------- small math helpers ----------
__device__ __forceinline__ int argmax_n(const float* v, int n) {
  float m = v[0]; int b = 0;
  for (int i = 1; i < n; ++i) { float x = v[i]; if (x > m) { m = x; b = i; } }
  return b;
}
__device__ __forceinline__ float nll_n(const float* v, int n, int lbl) {
  float m = v[0];
  for (int i = 1; i < n; ++i) m = fmaxf(m, v[i]);
  float s = 0.f;
  for (int i = 0; i < n; ++i) s += expf(v[i] - m);
  return logf(s) + m - v[lbl];
}
__device__ __forceinline__ float logsig(float x) {   // stable log(sigmoid(x))
  return fminf(x, 0.f) - log1pf(expf(-fabsf(x)));
}

#if USE_TDM
// Issue a 1-D TDM copy of TILE_ELEMS f32 from global to LDS (one wave issues it).
// amdgpu-toolchain clang-23 builtin arity: (u32x4, i32x8, i32x4, i32x4, i32x8, i32 cpol)
__device__ __forceinline__ void tdm_load_tile(const float* gsrc, float* ldst) {
  unsigned long long ga = (unsigned long long)(uintptr_t)gsrc;
  unsigned int laddr = (unsigned int)(uintptr_t)ldst;  // low 32 bits = LDS byte offset
  u32x4 g0;
  g0[0] = 1u;                                   // count=1 (valid), user mode
  g0[1] = laddr;                                // lds_addr (bytes)
  g0[2] = (unsigned int)ga;                     // global_addr[31:0]
  g0[3] = ((unsigned int)(ga >> 32) & 0x01FFFFFFu) | (2u << 30); // addr[56:32] | type=2
  i32x8 g1;
  g1[0] = (2 << 16);                            // data_size=4B; no mask/pad/iterate
  g1[1] = (int)((TILE_ELEMS & 0xFFFF) << 16);   // tensor_dim0[15:0] in [31:16]
  g1[2] = (int)(1u << 16);                      // tensor_dim0[31:16]=0 ; tensor_dim1=1
  g1[3] = (int)((unsigned)TILE_ELEMS << 16);    // tensor_dim1 hi=0 ; tile_dim0=4032
  g1[4] = (int)(1u | (1u << 16));               // tile_dim1=1 ; tile_dim2=1
  g1[5] = TILE_ELEMS;                           // tensor_dim0_stride lo32
  g1[6] = 0;                                    // stride0 hi ; stride1 lo16
  g1[7] = 0;
  i32x4 g2; g2[0] = 1; g2[1] = 1; g2[2] = 0; g2[3] = 0;  // dims2/3=1, tile_dim3=0
  i32x4 g3; g3[0] = 0; g3[1] = 0; g3[2] = 0; g3[3] = 0;
  i32x8 g4;                                     // extra group (zero-filled, probe-verified form)
  g4[0] = 0; g4[1] = 0; g4[2] = 0; g4[3] = 0;
  g4[4] = 0; g4[5] = 0; g4[6] = 0; g4[7] = 0;
  __builtin_amdgcn_tensor_load_to_lds(g0, g1, g2, g3, g4, 0);
}
#endif

// ---------- kernel 1: per-block partial reductions ----------
__global__ void __launch_bounds__(THREADS)
progloss_partials(const float* __restrict__ P, const float* __restrict__ T,
                  const float* __restrict__ W, float* __restrict__ ws) {
  __shared__ float ldsP[TILE_ELEMS];
  __shared__ float ldsT[TILE_ELEMS];
  __shared__ float ldsW[TILE_ELEMS];
  __shared__ float red[2][NACC];

  float acc[NACC];
#pragma unroll
  for (int j = 0; j < NACC; ++j) acc[j] = 0.f;
  acc[36] = BIGF;                               // min(first cub row)

  const int tid = threadIdx.x;
  for (int t = 0; t < TILES_PER_BLOCK; ++t) {
    const int tile = blockIdx.x * TILES_PER_BLOCK + t;
    const long long base = (long long)tile * TILE_ELEMS;

#if USE_TDM
    if (tid < 32) {                             // wave 0 drives the DMA
      tdm_load_tile(P + base, ldsP);
      tdm_load_tile(T + base, ldsT);
      tdm_load_tile(W + base, ldsW);
      __builtin_amdgcn_s_wait_tensorcnt(0);
    }
    __syncthreads();
#else
    for (int i = tid; i < TILE_ELEMS; i += THREADS) {
      ldsP[i] = P[base + i]; ldsT[i] = T[base + i]; ldsW[i] = W[base + i];
    }
    __syncthreads();
#endif

    // one row per thread; lds[tid*63 + d] is bank-conflict-free (gcd(63,64)=1)
    const int r = tile * ROWS_PER_TILE + tid;   // global row index
    const float* Pr = &ldsP[tid * D_COLS];
    const float* Tr = &ldsT[tid * D_COLS];
    const float* Wr = &ldsW[tid * D_COLS];

    const int cmd  = argmax_n(Tr, 7);
    const int pcmd = argmax_n(Pr, 7);
    acc[0] += nll_n(Pr, 7, cmd);                       // cmd_loss (mask = ones)
    if (cmd == pcmd) acc[1] += 1.f;                    // cmdc

    const int c1 = argmax_n(Tr + 7, 11),  pc1 = argmax_n(Pr + 7, 11);
    const int c2 = argmax_n(Tr + 18, 11), pc2 = argmax_n(Pr + 18, 11);
    const int c3 = argmax_n(Tr + 29, 11), pc3 = argmax_n(Pr + 29, 11);
    const int ax = argmax_n(Tr + 49, 3),  pax = argmax_n(Pr + 49, 3);
    const int fc = argmax_n(Tr + 54, 6),  pfc = argmax_n(Pr + 54, 6);

    float mae = 0.f;
    if (cmd >= 1 && cmd <= 5 && cmd != 3) {            // cub/ap/tr/sq need MAE
      for (int d = 0; d < D_COLS; ++d) mae += Wr[d] * fabsf(Pr[d] - Tr[d]);
    }

    if (cmd == 1) {                                    // cub
      acc[2] += mae; acc[26] += 1.f;
      if (r >= 1) acc[30] += 1.f;
      if (r >= 2) acc[31] += 1.f;
      acc[36] = fminf(acc[36], (float)r);
      const float t62 = Tr[62], x62 = Pr[62];
      acc[20] += -(POS_W * t62 * logsig(x62) + (1.f - t62) * logsig(-x62));
      if (t62 == 1.f) { acc[21] += 1.f; if (x62 > 0.f)  acc[23] += 1.f; }
      if (t62 == 0.f) { acc[22] += 1.f; if (x62 <= 0.f) acc[24] += 1.f; }
    } else if (cmd == 2) {                             // ap
      acc[3] += mae; acc[25] += 1.f; if (r >= 1) acc[32] += 1.f;
      acc[6] += nll_n(Pr + 7, 11, c1) + nll_n(Pr + 18, 11, c2);
      if (pc1 == c1 && pc2 == c2) acc[12] += 1.f;
    } else if (cmd == 3) {                             // ref
      acc[27] += 1.f; if (r >= 1) acc[33] += 1.f;
      acc[8]  += nll_n(Pr + 7, 11, c1);
      acc[10] += nll_n(Pr + 49, 3, ax);
      if (pc1 == c1) acc[14] += 1.f;
      if (pax == ax) acc[16] += 1.f;
    } else if (cmd == 4) {                             // tr
      acc[5] += mae; acc[28] += 1.f; if (r >= 1) acc[34] += 1.f;
      acc[9]  += nll_n(Pr + 7, 11, c1);
      acc[11] += nll_n(Pr + 49, 3, ax);
      if (pc1 == c1) acc[15] += 1.f;
      if (pax == ax) acc[17] += 1.f;
    } else if (cmd == 5) {                             // sq
      acc[4] += mae; acc[29] += 1.f; if (r >= 1) acc[35] += 1.f;
      acc[7] += nll_n(Pr + 7, 11, c1) + nll_n(Pr + 18, 11, c2) + nll_n(Pr + 29, 11, c3);
      if (pc1 == c1 && pc2 == c2 && pc3 == c3) acc[13] += 1.f;
      acc[19] += nll_n(Pr + 54, 6, fc);
      if (pfc == fc) acc[18] += 1.f;
    }
    __syncthreads();                                   // LDS reused next tile
  }

  // wave-level then block-level reduction
  const int lane = tid & 31, wid = tid >> 5;
#pragma unroll
  for (int j = 0; j < NACC; ++j) {
    float v = acc[j];
    if (j == 36) {
      for (int off = 16; off; off >>= 1) v = fminf(v, __shfl_down(v, off, 32));
    } else {
      for (int off = 16; off; off >>= 1) v += __shfl_down(v, off, 32);
    }
    if (lane == 0) red[wid][j] = v;
  }
  __syncthreads();
  if (tid == 0) {
    float* o = ws + (size_t)blockIdx.x * WS_STRIDE;
#pragma unroll
    for (int j = 0; j < NACC; ++j)
      o[j] = (j == 36) ? fminf(red[0][j], red[1][j]) : red[0][j] + red[1][j];
  }
}

// ---------- kernel 2: final reduce + first-row fixup + gating ----------
__global__ void __launch_bounds__(256)
progloss_finalize(const float* __restrict__ ws, const float* __restrict__ P,
                  const float* __restrict__ T, float* __restrict__ out) {
  __shared__ float red[8][NACC];
  float s[NACC];
#pragma unroll
  for (int j = 0; j < NACC; ++j) s[j] = 0.f;
  s[36] = BIGF;

  for (int i = threadIdx.x; i < NBLOCKS; i += 256) {
    const float* p = ws + (size_t)i * WS_STRIDE;
#pragma unroll
    for (int j = 0; j < NACC; ++j) {
      if (j == 36) s[j] = fminf(s[j], p[j]); else s[j] += p[j];
    }
  }
  const int lane = threadIdx.x & 31, wid = threadIdx.x >> 5;
#pragma unroll
  for (int j = 0; j < NACC; ++j) {
    float v = s[j];
    if (j == 36) {
      for (int off = 16; off; off >>= 1) v = fminf(v, __shfl_down(v, off, 32));
    } else {
      for (int off = 16; off; off >>= 1) v += __shfl_down(v, off, 32);
    }
    if (lane == 0) red[wid][j] = v;
  }
  __syncthreads();
  if (threadIdx.x == 0) {
    float f[NACC];
#pragma unroll
    for (int j = 0; j < NACC; ++j) {
      float v = red[0][j];
      for (int w = 1; w < 8; ++w) v = (j == 36) ? fminf(v, red[w][j]) : v + red[w][j];
      f[j] = v;
    }
    const bool g_cub = f[30] > 0.f, g_cub2 = f[31] > 0.f, g_ap = f[32] > 0.f;
    const bool g_ref = f[33] > 0.f, g_tr = f[34] > 0.f, g_sq = f[35] > 0.f;

    float alignL = 0.f, posn = 0.f, negn = 0.f, posc = 0.f, negc = 0.f;
    if (g_cub2) {                      // exclude first cub row from align sums
      const long long fr = (long long)f[36];
      const float x62 = P[fr * D_COLS + 62], t62 = T[fr * D_COLS + 62];
      const float b = -(POS_W * t62 * logsig(x62) + (1.f - t62) * logsig(-x62));
      alignL = f[20] - b;
      const bool pos = (t62 == 1.f), neg = (t62 == 0.f);
      posn = f[21] - (pos ? 1.f : 0.f);
      negn = f[22] - (neg ? 1.f : 0.f);
      posc = f[23] - ((pos && x62 > 0.f) ? 1.f : 0.f);
      negc = f[24] - ((neg && x62 <= 0.f) ? 1.f : 0.f);
    }
    out[0]  = f[0];
    out[1]  = g_cub ? f[2] : 0.f;
    out[2]  = g_ap  ? f[3] : 0.f;
    out[3]  = g_sq  ? f[4] : 0.f;
    out[4]  = g_tr  ? f[5] : 0.f;
    out[5]  = g_ap  ? f[6] : 0.f;
    out[6]  = g_sq  ? f[7] : 0.f;
    out[7]  = (g_ref ? f[8]  : 0.f) + (g_tr ? f[9]  : 0.f);
    out[8]  = (g_ref ? f[10] : 0.f) + (g_tr ? f[11] : 0.f);
    out[9]  = g_sq ? f[19] : 0.f;
    out[10] = f[1];
    out[11] = g_ap ? f[12] : 0.f;
    out[12] = g_sq ? f[13] : 0.f;
    out[13] = (g_ref ? f[14] : 0.f) + (g_tr ? f[15] : 0.f);
    out[14] = (g_ref ? f[16] : 0.f) + (g_tr ? f[17] : 0.f);
    out[15] = g_sq ? f[18] : 0.f;
    out[16] = alignL;
    out[17] = posc;
    out[18] = negc;
    out[19] = negn;
    out[20] = posn;
    out[21] = f[25];
    out[22] = f[26];
    out[23] = f[27] + f[28];
    out[24] = f[29];
  }
}

extern "C" void kernel_launch(void* const* d_in, const int* in_sizes, int n_in,
                              void* d_out, int out_size, void* d_ws, size_t ws_size,
                              hipStream_t stream) {
  const float* P = (const float*)d_in[0];
  const float* T = (const float*)d_in[1];
  const float* W = (const float*)d_in[2];
  float* ws = (float*)d_ws;
  progloss_partials<<<NBLOCKS, THREADS, 0, stream>>>(P, T, W, ws);
  progloss_finalize<<<1, 256, 0, stream>>>(ws, P, T, (float*)d_out);
}